// CausalGNNRecommender_66640712565003
// MI455X (gfx1250) — compile-verified
//
#include <hip/hip_runtime.h>
#include <hip/hip_bf16.h>

#define U_CNT 4096
#define I_CNT 12288
#define HD    64
#define N_NODES 16384
#define E_CNT 524288
#define NSPLIT 4
#define NT (N_NODES / 16)            // 1024 column tiles
#define TILES_PER_SPLIT (NT / NSPLIT)

typedef float v2f __attribute__((ext_vector_type(2)));
typedef float v4f __attribute__((ext_vector_type(4)));
typedef float v8f __attribute__((ext_vector_type(8)));

__device__ __forceinline__ v8f wmma_f32_4(v2f a, v2f b, v8f c) {
  // D = A(16x4 f32) * B(4x16 f32) + C(16x16 f32)
  return __builtin_amdgcn_wmma_f32_16x16x4_f32(false, a, false, b, (short)0, c,
                                               false, false);
}

// ---------------------------------------------------------------------------
// Kernel 1: xcat = concat(user,item); Q = xcat @ attn_w.T + attn_b
// ---------------------------------------------------------------------------
__global__ __launch_bounds__(256)
void build_q_kernel(const float* __restrict__ user_emb,
                    const float* __restrict__ item_emb,
                    const float* __restrict__ attn_w,
                    const float* __restrict__ attn_b,
                    float* __restrict__ xcat,
                    float* __restrict__ Q) {
  __shared__ float ws[HD * HD];
  __shared__ float xs[4][HD];
  const int tid = threadIdx.x;
#pragma unroll
  for (int i = 0; i < 16; ++i) ws[tid + 256 * i] = attn_w[tid + 256 * i];
  const int r = tid >> 6, h = tid & 63;
  const int row = blockIdx.x * 4 + r;
  float xv = (row < U_CNT) ? user_emb[row * HD + h]
                           : item_emb[(row - U_CNT) * HD + h];
  xs[r][h] = xv;
  xcat[row * HD + h] = xv;
  __syncthreads();
  float acc = attn_b[h];
#pragma unroll
  for (int k = 0; k < HD; ++k) acc += xs[r][k] * ws[h * HD + k];
  Q[row * HD + h] = acc;
}

// ---------------------------------------------------------------------------
// Pre-pack K^T (= causal_adj [64,N]) into WMMA-B-fragment order.
// Layout: adjB[jt][c2(8)][lane(32)][4] ; chunk 2*c2 -> {f0,f1}, 2*c2+1 -> {f2,f3}
// ---------------------------------------------------------------------------
__global__ __launch_bounds__(256)
void pack_k_kernel(const float* __restrict__ adj, float* __restrict__ adjB) {
  const unsigned idx = blockIdx.x * 256 + threadIdx.x;   // < N*64
  const int f    = idx & 3;
  const int lane = (idx >> 2) & 31;
  const int c2   = (idx >> 7) & 7;
  const int jt   = idx >> 10;
  const int hl = lane >> 4, nl = lane & 15;
  const int chunk = 2 * c2 + (f >> 1);
  const int k = 4 * chunk + 2 * hl + (f & 1);
  adjB[idx] = adj[(size_t)k * N_NODES + jt * 16 + nl];
}

// ---------------------------------------------------------------------------
// Pre-pack V (= xcat [N,64]) into WMMA-B-fragment order.
// Layout: Vb[jt][c(4)][lane(32)][8] ; f = 2*t + r -> V[jt*16+4c+2hl+r][16t+nl]
// ---------------------------------------------------------------------------
__global__ __launch_bounds__(256)
void pack_v_kernel(const float* __restrict__ xcat, float* __restrict__ Vb) {
  const unsigned idx = blockIdx.x * 256 + threadIdx.x;   // < N*64
  const int f    = idx & 7;
  const int lane = (idx >> 3) & 31;
  const int c    = (idx >> 8) & 3;
  const int jt   = idx >> 10;
  const int hl = lane >> 4, nl = lane & 15;
  const int j = jt * 16 + 4 * c + 2 * hl + (f & 1);
  const int h = 16 * (f >> 1) + nl;
  Vb[idx] = xcat[(size_t)j * HD + h];
}

// ---------------------------------------------------------------------------
// Kernel 2: split-K flash attention, fp32 WMMA 16x16x4.
// Softmax without max-subtraction (scores bounded ~|30|, fp32-safe):
// per tile only exp + per-lane l accumulation; row reduction once at the end.
// ---------------------------------------------------------------------------
__global__ __launch_bounds__(128)
void flash_attn_kernel(const float* __restrict__ Q,
                       const float* __restrict__ adjB,
                       const float* __restrict__ Vb,
                       float* __restrict__ Opart,
                       float* __restrict__ lpart) {
  __shared__ float pbuf[4][16 * 18];  // per-wave P tile, padded (stride 18)
  const int lane = threadIdx.x & 31;
  const int wave = threadIdx.x >> 5;
  const int nl = lane & 15;
  const int hl = lane >> 4;
  const int r0 = (blockIdx.x * 4 + wave) * 16;
  const int sp = blockIdx.y;
  float* pl = &pbuf[wave][0];

  // Q tile resident in A-fragment layout
  v2f qa[16];
#pragma unroll
  for (int c = 0; c < 16; ++c)
    qa[c] = *(const v2f*)(Q + (size_t)(r0 + nl) * HD + 4 * c + 2 * hl);

  v8f o0 = {}, o1 = {}, o2 = {}, o3 = {};
  v8f lrow = {};   // per-lane partial row sums (reduced after the loop)

  const int jt0 = sp * TILES_PER_SPLIT;
  const float* kb = adjB + (size_t)jt0 * 1024 + lane * 4;
  const float* vb = Vb   + (size_t)jt0 * 1024 + lane * 8;

  for (int it = 0; it < TILES_PER_SPLIT; ++it) {
    // ---- S = Q_tile @ K^T_tile ; dual accumulators halve the dep chain ----
    v8f sA = {}, sB = {};
#pragma unroll
    for (int c2 = 0; c2 < 4; ++c2) {
      v4f kw = *(const v4f*)(kb + c2 * 128);
      sA = wmma_f32_4(qa[2 * c2],     kw.xy, sA);
      sA = wmma_f32_4(qa[2 * c2 + 1], kw.zw, sA);
    }
#pragma unroll
    for (int c2 = 4; c2 < 8; ++c2) {
      v4f kw = *(const v4f*)(kb + c2 * 128);
      sB = wmma_f32_4(qa[2 * c2],     kw.xy, sB);
      sB = wmma_f32_4(qa[2 * c2 + 1], kw.zw, sB);
    }
    v8f s = sA + sB;

    // ---- P = exp(S); accumulate per-lane row-sum; stage P for transpose ----
#pragma unroll
    for (int v = 0; v < 8; ++v) {
      const float p = __expf(s[v]);
      lrow[v] += p;
      pl[(v + 8 * hl) * 18 + nl] = p;  // C-layout -> row-major LDS
    }
    // ---- reload P as A-fragments ----
    v2f pa[4];
#pragma unroll
    for (int c = 0; c < 4; ++c)
      pa[c] = *(const v2f*)&pl[nl * 18 + 4 * c + 2 * hl];
    // ---- O += P @ V_tile : ILP-4 across h-chunks ----
#pragma unroll
    for (int c = 0; c < 4; ++c) {
      v4f w0 = *(const v4f*)(vb + c * 256);
      v4f w1 = *(const v4f*)(vb + c * 256 + 4);
      o0 = wmma_f32_4(pa[c], w0.xy, o0);
      o1 = wmma_f32_4(pa[c], w0.zw, o1);
      o2 = wmma_f32_4(pa[c], w1.xy, o2);
      o3 = wmma_f32_4(pa[c], w1.zw, o3);
    }
    kb += 1024;
    vb += 1024;
  }

  // ---- one-time row-sum reduction across the 16-lane group ----
#pragma unroll
  for (int v = 0; v < 8; ++v) {
    float rs = lrow[v];
#pragma unroll
    for (int m = 1; m < 16; m <<= 1) rs += __shfl_xor(rs, m);
    lrow[v] = rs;
  }

  // ---- store unnormalized partials ----
  float* op = Opart + (size_t)sp * N_NODES * HD;
#pragma unroll
  for (int v = 0; v < 8; ++v) {
    const int row = r0 + v + 8 * hl;
    const size_t ro = (size_t)row * HD;
    op[ro + 0 + nl]  = o0[v];
    op[ro + 16 + nl] = o1[v];
    op[ro + 32 + nl] = o2[v];
    op[ro + 48 + nl] = o3[v];
    if (nl == 0) lpart[(size_t)sp * N_NODES + row] = lrow[v];
  }
}

// ---------------------------------------------------------------------------
// Combine split partials: att = (sum_s O_s) / (sum_s l_s)
// ---------------------------------------------------------------------------
__global__ __launch_bounds__(256)
void combine_kernel(const float* __restrict__ Opart,
                    const float* __restrict__ lpart,
                    float* __restrict__ att) {
  const unsigned idx = blockIdx.x * 256 + threadIdx.x;  // < N*64
  const int row = idx >> 6;
  float L = 0.0f, O = 0.0f;
#pragma unroll
  for (int s = 0; s < NSPLIT; ++s) {
    L += lpart[(size_t)s * N_NODES + row];
    O += Opart[(size_t)s * N_NODES * HD + idx];
  }
  att[idx] = O / L;
}

// ---------------------------------------------------------------------------
// SAGE conv pieces
// ---------------------------------------------------------------------------
__global__ void deg_kernel(const int* __restrict__ tgt, float* __restrict__ deg) {
  const int e = blockIdx.x * blockDim.x + threadIdx.x;
  if (e < E_CNT) atomicAdd(&deg[tgt[e]], 1.0f);
}

__global__ void scatter_kernel(const int* __restrict__ src,
                               const int* __restrict__ tgt,
                               const float* __restrict__ xin,
                               float* __restrict__ aggr) {
  const unsigned idx = blockIdx.x * blockDim.x + threadIdx.x;
  const unsigned e = idx >> 6, h = idx & 63;
  atomicAdd(&aggr[(size_t)tgt[e] * HD + h], xin[(size_t)src[e] * HD + h]);
}

__global__ __launch_bounds__(256)
void sage_linear_kernel(const float* __restrict__ xin,
                        const float* __restrict__ aggr,
                        const float* __restrict__ deg,
                        const float* __restrict__ lw,
                        const float* __restrict__ lb,
                        const float* __restrict__ rw,
                        float* __restrict__ out) {
  __shared__ float lws[HD * HD];
  __shared__ float rws[HD * HD];
  __shared__ float xs[4][HD];
  __shared__ float as[4][HD];
  const int tid = threadIdx.x;
#pragma unroll
  for (int i = 0; i < 16; ++i) {
    lws[tid + 256 * i] = lw[tid + 256 * i];
    rws[tid + 256 * i] = rw[tid + 256 * i];
  }
  const int r = tid >> 6, h = tid & 63;
  const int row = blockIdx.x * 4 + r;
  float d = deg[row];
  d = d < 1.0f ? 1.0f : d;
  xs[r][h] = xin[(size_t)row * HD + h];
  as[r][h] = aggr[(size_t)row * HD + h] * (1.0f / d);
  __syncthreads();
  float acc = lb[h];
#pragma unroll
  for (int k = 0; k < HD; ++k)
    acc += as[r][k] * lws[h * HD + k] + xs[r][k] * rws[h * HD + k];
  out[(size_t)row * HD + h] = fmaxf(acc, 0.0f);
}

// ---------------------------------------------------------------------------
extern "C" void kernel_launch(void* const* d_in, const int* in_sizes, int n_in,
                              void* d_out, int out_size, void* d_ws, size_t ws_size,
                              hipStream_t stream) {
  const int*   ei       = (const int*)d_in[0];
  const float* user_emb = (const float*)d_in[1];
  const float* item_emb = (const float*)d_in[2];
  const float* attn_w   = (const float*)d_in[3];
  const float* attn_b   = (const float*)d_in[4];
  const float* adj      = (const float*)d_in[5];
  const float* l0_lw    = (const float*)d_in[6];
  const float* l0_lb    = (const float*)d_in[7];
  const float* l0_rw    = (const float*)d_in[8];
  const float* l1_lw    = (const float*)d_in[9];
  const float* l1_lb    = (const float*)d_in[10];
  const float* l1_rw    = (const float*)d_in[11];
  float* out = (float*)d_out;

  const size_t NH = (size_t)N_NODES * HD;
  float* ws    = (float*)d_ws;
  float* xcat  = ws;
  float* Q     = ws + 1 * NH;
  float* att   = ws + 2 * NH;
  float* adjB  = ws + 3 * NH;
  float* Vb    = ws + 4 * NH;
  float* Opart = ws + 5 * NH;            // NSPLIT * NH
  float* aggr  = Opart;                  // reused after combine
  float* h1    = Opart + NH;             // reused after combine
  float* lpart = ws + (5 + NSPLIT) * NH;
  float* deg   = lpart + (size_t)NSPLIT * N_NODES;

  const int* src = ei;
  const int* tgt = ei + E_CNT;

  build_q_kernel<<<N_NODES / 4, 256, 0, stream>>>(user_emb, item_emb, attn_w,
                                                  attn_b, xcat, Q);
  pack_k_kernel<<<(N_NODES * HD) / 256, 256, 0, stream>>>(adj, adjB);
  pack_v_kernel<<<(N_NODES * HD) / 256, 256, 0, stream>>>(xcat, Vb);

  dim3 fgrid(N_NODES / 64, NSPLIT);
  flash_attn_kernel<<<fgrid, 128, 0, stream>>>(Q, adjB, Vb, Opart, lpart);
  combine_kernel<<<(N_NODES * HD) / 256, 256, 0, stream>>>(Opart, lpart, att);

  hipMemsetAsync(deg, 0, N_NODES * sizeof(float), stream);
  deg_kernel<<<(E_CNT + 255) / 256, 256, 0, stream>>>(tgt, deg);

  // Layer 0
  hipMemsetAsync(aggr, 0, NH * sizeof(float), stream);
  scatter_kernel<<<(E_CNT * 64) / 256, 256, 0, stream>>>(src, tgt, att, aggr);
  sage_linear_kernel<<<N_NODES / 4, 256, 0, stream>>>(att, aggr, deg, l0_lw,
                                                      l0_lb, l0_rw, h1);
  // Layer 1 -> d_out
  hipMemsetAsync(aggr, 0, NH * sizeof(float), stream);
  scatter_kernel<<<(E_CNT * 64) / 256, 256, 0, stream>>>(src, tgt, h1, aggr);
  sage_linear_kernel<<<N_NODES / 4, 256, 0, stream>>>(h1, aggr, deg, l1_lw,
                                                      l1_lb, l1_rw, out);
}